// SumConv_31559419691540
// MI455X (gfx1250) — compile-verified
//
#include <hip/hip_runtime.h>

typedef float v2f __attribute__((ext_vector_type(2)));
typedef float v4f __attribute__((ext_vector_type(4)));
typedef float v8f __attribute__((ext_vector_type(8)));

#define B_    128
#define F_    1024
#define CIN_  16
#define COUT_ 16
#define K_    4
#define R_    2

// out[b,f,o,r] = m + log( sum_cin softmax_cin(logits)[o,cin,kh,kw,r] * exp(ll[b,f,cin,r]-m) )
// One wave32 handles (b, kh, kw, 16 spatial positions) for BOTH r values:
//   - ll loads   : 4 x b128 per lane, every byte consumed
//   - out stores : 8 x b64 per lane, each 128B (o,r) block fully tiled by lanes l / l+16
//   - math       : 2 accumulators, 8 chained V_WMMA_F32_16X16X4_F32 (K=16 each, exact fp32)
__global__ __launch_bounds__(256) void sumconv_wmma_kernel(
    const float* __restrict__ ll,      // (B, F, CIN, R)
    const float* __restrict__ logits,  // (COUT, CIN, K, K, R)
    float* __restrict__ out)           // (B, F, COUT, R)
{
  const int wave = (blockIdx.x * (int)blockDim.x + (int)threadIdx.x) >> 5;
  const int lane = threadIdx.x & 31;
  const int hi   = lane >> 4;    // 0: K-subcols {0,1}; 1: K-subcols {2,3} of each K=4 chunk
  const int col  = lane & 15;    // A: row M ; B/D: column N

  // wave -> (b, kh, kw, group)   [8192 waves]
  const int g  =  wave       & 3;    // group of 16 positions (of 64 sharing (kh,kw))
  const int kw = (wave >> 2) & 3;
  const int kh = (wave >> 4) & 3;
  const int b  =  wave >> 6;         // 0..127

  // lane slot j (0..7) handles cin_j = 4*(j>>1) + 2*hi + (j&1)

  // ---------------- A fragments: row-softmax of logits[o=col, :, kh, kw, r], r=0,1 ----
  float lg0[8], lg1[8];
  const int lbase = col * (CIN_ * K_ * K_ * R_) + kh * (K_ * R_) + kw * R_;  // + cin*32 (+r)
  #pragma unroll
  for (int j = 0; j < 8; ++j) {
    const int cin = 4 * (j >> 1) + 2 * hi + (j & 1);
    const v2f p = *(const v2f*)(logits + lbase + cin * (K_ * K_ * R_));  // (r0, r1)
    lg0[j] = p.x; lg1[j] = p.y;
  }
  float wm0 = lg0[0], wm1 = lg1[0];
  #pragma unroll
  for (int j = 1; j < 8; ++j) { wm0 = fmaxf(wm0, lg0[j]); wm1 = fmaxf(wm1, lg1[j]); }
  wm0 = fmaxf(wm0, __shfl_xor(wm0, 16, 32));          // full row max over 16 cin
  wm1 = fmaxf(wm1, __shfl_xor(wm1, 16, 32));
  float ae0[8], ae1[8], ps0 = 0.0f, ps1 = 0.0f;
  #pragma unroll
  for (int j = 0; j < 8; ++j) {
    ae0[j] = __expf(lg0[j] - wm0); ps0 += ae0[j];
    ae1[j] = __expf(lg1[j] - wm1); ps1 += ae1[j];
  }
  const float wi0 = 1.0f / (ps0 + __shfl_xor(ps0, 16, 32));
  const float wi1 = 1.0f / (ps1 + __shfl_xor(ps1, 16, 32));
  v2f af0[4], af1[4];
  #pragma unroll
  for (int c = 0; c < 4; ++c) {
    af0[c].x = ae0[2 * c] * wi0; af0[c].y = ae0[2 * c + 1] * wi0;
    af1[c].x = ae1[2 * c] * wi1; af1[c].y = ae1[2 * c + 1] * wi1;
  }

  // ---------------- B fragments: exp(ll - m) for this lane's column, r=0,1 ----------
  const int idx = g * 16 + col;                 // 0..63 within (kh,kw)
  const int hb  = idx >> 3, wb = idx & 7;
  const int f   = (4 * hb + kh) * 32 + 4 * wb + kw;
  const v4f* px = (const v4f*)(ll + (size_t)(b * F_ + f) * (CIN_ * R_)); // 128B block
  float x0[8], x1[8];
  #pragma unroll
  for (int c = 0; c < 4; ++c) {
    const v4f q = px[2 * c + hi];   // = ll[cin..cin+1][r0,r1], cin = 4c + 2*hi
    x0[2 * c]     = q.x; x1[2 * c]     = q.y;
    x0[2 * c + 1] = q.z; x1[2 * c + 1] = q.w;
  }
  float m0 = x0[0], m1 = x1[0];
  #pragma unroll
  for (int j = 1; j < 8; ++j) { m0 = fmaxf(m0, x0[j]); m1 = fmaxf(m1, x1[j]); }
  m0 = fmaxf(m0, __shfl_xor(m0, 16, 32));             // column max over 16 cin
  m1 = fmaxf(m1, __shfl_xor(m1, 16, 32));
  v2f bf0[4], bf1[4];
  #pragma unroll
  for (int c = 0; c < 4; ++c) {
    bf0[c].x = __expf(x0[2 * c]     - m0); bf0[c].y = __expf(x0[2 * c + 1] - m0);
    bf1[c].x = __expf(x1[2 * c]     - m1); bf1[c].y = __expf(x1[2 * c + 1] - m1);
  }

  // ---------------- D = W x E  (fp32 WMMA, K=16 via 4 chained K=4 steps, x2 for r) ----
  v8f acc0 = {0.f, 0.f, 0.f, 0.f, 0.f, 0.f, 0.f, 0.f};
  v8f acc1 = {0.f, 0.f, 0.f, 0.f, 0.f, 0.f, 0.f, 0.f};
  #pragma unroll
  for (int c = 0; c < 4; ++c) {
    acc0 = __builtin_amdgcn_wmma_f32_16x16x4_f32(false, af0[c], false, bf0[c],
                                                 (short)0, acc0, false, false);
    acc1 = __builtin_amdgcn_wmma_f32_16x16x4_f32(false, af1[c], false, bf1[c],
                                                 (short)0, acc1, false, false);
  }

  // ---------------- epilogue: out[...,o,r] = m_r + log(acc_r[o]) as b64 stores -------
  v2f* po = (v2f*)(out + (size_t)(b * F_ + f) * (COUT_ * R_));
  #pragma unroll
  for (int v = 0; v < 8; ++v) {
    const int o = hi * 8 + v;                   // D VGPR v holds M = v + 8*hi
    v2f w; w.x = m0 + __logf(acc0[v]); w.y = m1 + __logf(acc1[v]);
    po[o] = w;
  }
}

extern "C" void kernel_launch(void* const* d_in, const int* in_sizes, int n_in,
                              void* d_out, int out_size, void* d_ws, size_t ws_size,
                              hipStream_t stream) {
  (void)in_sizes; (void)n_in; (void)d_ws; (void)ws_size; (void)out_size;
  const float* ll     = (const float*)d_in[0];
  const float* logits = (const float*)d_in[1];
  float* out          = (float*)d_out;

  // waves = B * (K*K) * 4 groups = 8192 ; 8 waves/block -> 1024 blocks
  dim3 grid(1024), block(256);
  sumconv_wmma_kernel<<<grid, block, 0, stream>>>(ll, logits, out);
}